// GradientInputLayer_25391846654370
// MI455X (gfx1250) — compile-verified
//
#include <hip/hip_runtime.h>
#include <hip/hip_bf16.h>
#include <stdint.h>

// Problem constants
#define BDIM   64
#define CDIM   2
#define LDIM   64
#define NBINS  16
#define NV     (LDIM*LDIM*LDIM)        // 262144 voxels
#define NROWS  (BDIM*CDIM)             // 128 (b,c) rows
#define NB2    (NBINS*NBINS)           // 256 bins
#define EPSV   1e-5f
#define QSPLIT 4                       // voxel-range split per row (occupancy)
#define NPART  (NROWS*QSPLIT)          // 512 partial histograms
#define WAVE_VOX (NV/QSPLIT/8)         // 8192 voxels per wave

typedef __attribute__((ext_vector_type(2))) float v2f;
typedef __attribute__((ext_vector_type(8))) float v8f;
typedef int v4i __attribute__((vector_size(16)));

#if __has_builtin(__builtin_amdgcn_global_load_async_to_lds_b128) && \
    __has_builtin(__builtin_amdgcn_s_wait_asynccnt)
#define USE_ASYNC 1
typedef __attribute__((address_space(3))) v4i lds_v4i;
#else
#define USE_ASYNC 0
#endif

// ---------------------------------------------------------------------------
// K1: per-voxel tables: mag (f32) and bin id (u8). 262144 voxels.
// ---------------------------------------------------------------------------
__global__ __launch_bounds__(256) void k_tables(float* __restrict__ magT,
                                                unsigned char* __restrict__ binT) {
    int v = blockIdx.x * 256 + threadIdx.x;
    int i = v >> 12;
    int j = (v >> 6) & 63;
    int k = v & 63;
    float fi = (float)i, fj = (float)j, fk = (float)k;
    float mag = sqrtf(fi*fi + fj*fj + fk*fk);
    float xc  = sqrtf(fj*fj + fk*fk);
    float zc  = sqrtf(fj*fj + fi*fi);
    const float s = 57.29577951308232f / 22.0f;   // (180/pi)/BIN, BIN = 360//16 = 22
    int ax = (int)floorf(atan2f(xc, fi) * s);
    int az = (int)floorf(atan2f(fk, zc) * s);
    magT[v] = mag;
    binT[v] = (unsigned char)(ax * NBINS + az);
}

// ---------------------------------------------------------------------------
// K2: weighted histogram. Block = (row, quarter); 8 waves, each with a
// private 256-bin LDS accumulator (ds_add_f32). x in {0,1} -> predicated
// atomics. x stream staged through LDS with async-to-LDS double buffering
// (ASYNCcnt) when the builtins exist; mag/bin tables are L2-hot direct loads.
// Writes deterministic per-block partials (no global atomics).
// ---------------------------------------------------------------------------
__global__ __launch_bounds__(256) void k_hist(const float* __restrict__ x,
                                              const float* __restrict__ magT,
                                              const unsigned char* __restrict__ binT,
                                              float* __restrict__ gpart) {
    __shared__ float wb[8][NB2];          // per-wave private bins (8 KB)
#if USE_ASYNC
    __shared__ float xstage[8][2][128];   // per-wave double buffer (8 KB)
#endif
    int tid = threadIdx.x;
    for (int b = tid; b < 8 * NB2; b += 256) ((float*)wb)[b] = 0.0f;
    __syncthreads();

    int row  = blockIdx.x >> 2;               // (b*2 + c)
    int q    = blockIdx.x & 3;
    int w    = tid >> 5;
    int lane = tid & 31;
    const float* xr = x + (size_t)row * NV;
    int base = q * (NV / QSPLIT) + w * WAVE_VOX;
    float* mybins = wb[w];
    const int iters = WAVE_VOX / 128;         // 64

#if USE_ASYNC
    lds_v4i* dstA = (lds_v4i*)&xstage[w][0][lane * 4];
    lds_v4i* dstB = (lds_v4i*)&xstage[w][1][lane * 4];
    // prologue: fill buffer 0
    __builtin_amdgcn_global_load_async_to_lds_b128(
        (v4i*)(xr + base + lane * 4), dstA, 0, 0);
    for (int t = 0; t < iters; ++t) {
        int off = base + t * 128 + lane * 4;
        __builtin_prefetch(magT + off + 1024, 0, 1);   // keep table ahead in L2
        if (t + 1 < iters) {
            __builtin_amdgcn_global_load_async_to_lds_b128(
                (v4i*)(xr + off + 128), (t & 1) ? dstA : dstB, 0, 0);
            __builtin_amdgcn_s_wait_asynccnt(1);       // buffer for t is ready
        } else {
            __builtin_amdgcn_s_wait_asynccnt(0);
        }
        __asm__ volatile("" ::: "memory");             // no reordering across wait
        float4 xv = *(const float4*)&xstage[w][t & 1][lane * 4];
        float4 mv = *(const float4*)(magT + off);
        unsigned int bw = *(const unsigned int*)(binT + off);
        if (xv.x != 0.0f) atomicAdd(&mybins[bw & 255], mv.x);
        if (xv.y != 0.0f) atomicAdd(&mybins[(bw >> 8) & 255], mv.y);
        if (xv.z != 0.0f) atomicAdd(&mybins[(bw >> 16) & 255], mv.z);
        if (xv.w != 0.0f) atomicAdd(&mybins[bw >> 24], mv.w);
    }
#else
    #pragma unroll 2
    for (int t = 0; t < iters; ++t) {
        int off = base + t * 128 + lane * 4;
        __builtin_prefetch(xr + off + 1024, 0, 1);     // global_prefetch_b8
        float4 xv = *(const float4*)(xr + off);
        float4 mv = *(const float4*)(magT + off);
        unsigned int bw = *(const unsigned int*)(binT + off);
        if (xv.x != 0.0f) atomicAdd(&mybins[bw & 255], mv.x);
        if (xv.y != 0.0f) atomicAdd(&mybins[(bw >> 8) & 255], mv.y);
        if (xv.z != 0.0f) atomicAdd(&mybins[(bw >> 16) & 255], mv.z);
        if (xv.w != 0.0f) atomicAdd(&mybins[bw >> 24], mv.w);
    }
#endif
    __syncthreads();

    // combine the 8 wave-private arrays -> gpart[block][256]
    float s = 0.0f;
    #pragma unroll
    for (int ww = 0; ww < 8; ++ww) s += wb[ww][tid];
    gpart[blockIdx.x * NB2 + tid] = s;
}

// ---------------------------------------------------------------------------
// K3: combine partials into g (held wholly in LDS: 128 KB of the 320 KB WGP
// LDS), per-channel mean/var over 16384 values via WMMA (A=ones 16x4,
// B=data 4x16, exact f32), then normalize + affine. Single block, 8 waves.
// ---------------------------------------------------------------------------
__global__ __launch_bounds__(256) void k_norm(const float* __restrict__ gpart,
                                              const float* __restrict__ gamma,
                                              const float* __restrict__ beta,
                                              float* __restrict__ out) {
    __shared__ float gs[NROWS * NB2];     // 128 KB — fits CDNA5 320 KB WGP LDS
    __shared__ float pS[8], pQ[8], meanS[2], invS[2];
    int tid = threadIdx.x;
    int w = tid >> 5, lane = tid & 31;
    int ch = w >> 2, p = w & 3;

    // phase 0: deterministic 4-way combine of partials
    for (int idx = tid; idx < NROWS * NB2; idx += 256) {
        int r = idx >> 8, col = idx & 255;
        float s = 0.0f;
        #pragma unroll
        for (int qq = 0; qq < QSPLIT; ++qq)
            s += gpart[(r * QSPLIT + qq) * NB2 + col];
        gs[idx] = s;
    }
    __syncthreads();

    // phase 1: per-channel sum / sumsq
    float sum, sq;
#if __has_builtin(__builtin_amdgcn_wmma_f32_16x16x4_f32)
    v2f ones = {1.0f, 1.0f};
    v8f accS = {}; v8f accQ = {};
    for (int t = 0; t < 16; ++t) {
        const float* rp = gs + (((p * 16 + t) * 2 + ch) * NB2);
        #pragma unroll
        for (int blk = 0; blk < 4; ++blk) {
            float e0 = rp[blk * 64 + lane];
            float e1 = rp[blk * 64 + 32 + lane];
            v2f bv = {e0, e1};
            v2f bq = {e0 * e0, e1 * e1};
            // D = ones(16x4) x B(4x16) + C : every D row = per-column sums.
            accS = __builtin_amdgcn_wmma_f32_16x16x4_f32(
                false, ones, false, bv, (short)0, accS, false, false);
            accQ = __builtin_amdgcn_wmma_f32_16x16x4_f32(
                false, ones, false, bq, (short)0, accQ, false, false);
        }
    }
    // C VGPR0 holds identical rows (M=0 lanes 0-15, M=8 lanes 16-31), so
    // lane-sum of accS[0] == 2 * total.
    sum = accS[0] * 0.5f;
    sq  = accQ[0] * 0.5f;
#else
    sum = 0.0f; sq = 0.0f;
    for (int t = 0; t < 16; ++t) {
        const float* rp = gs + (((p * 16 + t) * 2 + ch) * NB2);
        for (int blk = 0; blk < 4; ++blk) {
            float e0 = rp[blk * 64 + lane];
            float e1 = rp[blk * 64 + 32 + lane];
            sum += e0 + e1;
            sq  += e0 * e0 + e1 * e1;
        }
    }
#endif
    // wave32 reduction
    for (int o = 16; o > 0; o >>= 1) {
        sum += __shfl_down(sum, o, 32);
        sq  += __shfl_down(sq,  o, 32);
    }
    if (lane == 0) { pS[w] = sum; pQ[w] = sq; }
    __syncthreads();

    if (tid < 2) {
        float S = 0.0f, Q = 0.0f;
        #pragma unroll
        for (int qq = 0; qq < 4; ++qq) { S += pS[tid * 4 + qq]; Q += pQ[tid * 4 + qq]; }
        float m   = S * (1.0f / 16384.0f);
        float var = Q * (1.0f / 16384.0f) - m * m;
        meanS[tid] = m;
        invS[tid]  = rsqrtf(var + EPSV);
    }
    __syncthreads();

    float ga[2] = {gamma[0], gamma[1]};
    float be[2] = {beta[0],  beta[1]};
    for (int idx = tid; idx < NROWS * NB2; idx += 256) {
        int c = (idx >> 8) & 1;                      // row = idx>>8 = b*2+c
        out[idx] = (gs[idx] - meanS[c]) * invS[c] * ga[c] + be[c];
    }
}

// ---------------------------------------------------------------------------
extern "C" void kernel_launch(void* const* d_in, const int* in_sizes, int n_in,
                              void* d_out, int out_size, void* d_ws, size_t ws_size,
                              hipStream_t stream) {
    const float* x     = (const float*)d_in[0];
    const float* gamma = (const float*)d_in[1];
    const float* beta  = (const float*)d_in[2];
    float* out = (float*)d_out;

    // workspace layout: magT (1MB) | binT (256KB) | gpart (512KB)
    float* magT         = (float*)d_ws;
    unsigned char* binT = (unsigned char*)d_ws + (size_t)NV * 4;
    float* gpart        = (float*)((char*)d_ws + (size_t)NV * 4 + (size_t)NV);

    k_tables<<<NV / 256, 256, 0, stream>>>(magT, binT);
    k_hist  <<<NPART,    256, 0, stream>>>(x, magT, binT, gpart);
    k_norm  <<<1,        256, 0, stream>>>(gpart, gamma, beta, out);
}